// PagedAttentionModel_3410204033315
// MI455X (gfx1250) — compile-verified
//
#include <hip/hip_runtime.h>
#include <math.h>

#define NH    32
#define NKV   8
#define HD    64
#define GQA   4
#define HDIM  2048
#define INTER 8192
#define BATCH 32
#define NBLK  256
#define BS    16
#define EPSV  1e-5f

typedef float v2f __attribute__((ext_vector_type(2)));
typedef float v8f __attribute__((ext_vector_type(8)));

// ---------------------------------------------------------------------------
// CDNA5 async global->LDS path (guarded; falls back to load+ds_store).
// Prototype discovered from compiler diagnostics:
//   void __builtin_amdgcn_global_load_async_to_lds_b128(
//        v4i __device__* gsrc, v4i __shared__* ldst, int offset, int cpol)
// ---------------------------------------------------------------------------
#if defined(__has_builtin)
#if __has_builtin(__builtin_amdgcn_global_load_async_to_lds_b128) && \
    __has_builtin(__builtin_amdgcn_s_wait_asynccnt)
#define USE_ASYNC_LDS 1
#endif
#endif

#ifdef USE_ASYNC_LDS
typedef int v4i_t __attribute__((vector_size(16)));
typedef __attribute__((address_space(1))) v4i_t g_v4i;   // global
typedef __attribute__((address_space(3))) v4i_t s_v4i;   // LDS
#endif

// Stage a 32x16 f32 tile of W (row-major, row stride ldw) into LDS buf,
// block-cooperatively (64 threads, 128 b128 slots -> 2 per thread).
static __device__ __forceinline__
void stage_tile(const float* __restrict__ Wn0, size_t ldw, int k0,
                float* __restrict__ buf, int tid) {
  #pragma unroll
  for (int s = tid; s < 128; s += 64) {
    const int r = s >> 2, cg = (s & 3) << 2;
    const float* g = Wn0 + (size_t)(k0 + r) * ldw + cg;
    float* l = buf + r * 16 + cg;
#ifdef USE_ASYNC_LDS
    __builtin_amdgcn_global_load_async_to_lds_b128(
        (g_v4i*)g, (s_v4i*)l, 0, 0);
#else
    *(float4*)l = *(const float4*)g;
#endif
  }
}

// ---------------------------------------------------------------------------
// 16x16 f32 WMMA GEMM tile with LDS-staged, double-buffered B and dual
// accumulators.  arow = &A[m_lane][2*hi] (lane's row).  wtile = 2*512 floats.
// Layout per ISA 7.12.2: A lane m=lane%16 holds k={2hi,2hi+1};
// B lane n=lane%16, vgpr v holds k=v+2hi; C vgpr v -> m=v+8hi, n=lane%16.
// ---------------------------------------------------------------------------
static __device__ __forceinline__
v8f gemm_tile_lds(const float* __restrict__ arow,
                  const float* __restrict__ Wn0,
                  int K, size_t ldw, float* wtile,
                  int tid, int lo, int hi) {
  v8f c0 = {}, c1 = {};
  stage_tile(Wn0, ldw, 0, wtile, tid);
  for (int k0 = 0; k0 < K; k0 += 32) {
    float* cur = wtile + (((k0 >> 5) & 1) << 9);
    float* nxt = wtile + ((((k0 >> 5) + 1) & 1) << 9);
    if (k0 + 32 < K) {
      stage_tile(Wn0, ldw, k0 + 32, nxt, tid);
#ifdef USE_ASYNC_LDS
      __builtin_amdgcn_s_wait_asynccnt(2);   // cur's loads (older) complete
#endif
    } else {
#ifdef USE_ASYNC_LDS
      __builtin_amdgcn_s_wait_asynccnt(0);
#endif
    }
    __syncthreads();
    #pragma unroll
    for (int kk = 0; kk < 32; kk += 4) {
      v2f a = { arow[k0 + kk], arow[k0 + kk + 1] };
      v2f b = { cur[(kk + 2 * hi) * 16 + lo],
                cur[(kk + 2 * hi + 1) * 16 + lo] };
      if ((kk >> 2) & 1)
        c1 = __builtin_amdgcn_wmma_f32_16x16x4_f32(false, a, false, b,
                                                   (short)0, c1, false, false);
      else
        c0 = __builtin_amdgcn_wmma_f32_16x16x4_f32(false, a, false, b,
                                                   (short)0, c0, false, false);
    }
    __syncthreads();
  }
  return c0 + c1;
}

// ---------------------------------------------------------------------------
// RMSNorm (one block per batch row)
// ---------------------------------------------------------------------------
__global__ void rms_kernel(const float* __restrict__ x,
                           const float* __restrict__ w,
                           float* __restrict__ out) {
  __shared__ float red[256];
  const int b = blockIdx.x, t = threadIdx.x;
  const float* xr = x + (size_t)b * HDIM;
  float ss = 0.f;
  for (int i = t; i < HDIM; i += 256) { float v = xr[i]; ss += v * v; }
  red[t] = ss; __syncthreads();
  for (int off = 128; off > 0; off >>= 1) {
    if (t < off) red[t] += red[t + off];
    __syncthreads();
  }
  const float inv = rsqrtf(red[0] * (1.0f / HDIM) + EPSV);
  float* orow = out + (size_t)b * HDIM;
  for (int i = t; i < HDIM; i += 256) orow[i] = xr[i] * inv * w[i];
}

// ---------------------------------------------------------------------------
// Fused QKV projection. grid = 192 N-tiles (128 q / 32 k / 32 v), 64 threads.
// ---------------------------------------------------------------------------
__global__ void qkv_kernel(const float* __restrict__ normed,
                           const float* __restrict__ qw,
                           const float* __restrict__ kw,
                           const float* __restrict__ vw,
                           float* __restrict__ qo,
                           float* __restrict__ ko,
                           float* __restrict__ vo) {
  __shared__ float wtile[2 * 512];
  const int tid = threadIdx.x;
  const int wave = tid >> 5, lane = tid & 31;
  const int lo = lane & 15, hi = lane >> 4;
  const int bx = blockIdx.x;
  const float* W; float* out; size_t ldw; int n0;
  if (bx < 128)      { W = qw; out = qo; ldw = NH * HD;  n0 = bx * 16; }
  else if (bx < 160) { W = kw; out = ko; ldw = NKV * HD; n0 = (bx - 128) * 16; }
  else               { W = vw; out = vo; ldw = NKV * HD; n0 = (bx - 160) * 16; }

  const float* arow = normed + (size_t)(wave * 16 + lo) * HDIM + 2 * hi;
  v8f c = gemm_tile_lds(arow, W + n0, HDIM, ldw, wtile, tid, lo, hi);
  #pragma unroll
  for (int v = 0; v < 8; ++v) {
    const int m = wave * 16 + v + 8 * hi;
    out[(size_t)m * ldw + n0 + lo] = c[v];
  }
}

// ---------------------------------------------------------------------------
// RoPE on q and k_new, in place (pair (i, i+32) handled by one thread)
// ---------------------------------------------------------------------------
__global__ void rope_kernel(float* __restrict__ q, float* __restrict__ k,
                            const int* __restrict__ seqlens) {
  const int b = blockIdx.x, t = threadIdx.x;
  const float pos = (float)seqlens[b];
  for (int p = t; p < (NH + NKV) * (HD / 2); p += 256) {
    float* base; int head, i;
    if (p < NH * (HD / 2)) { base = q + (size_t)b * NH * HD;  head = p >> 5; i = p & 31; }
    else { int pp = p - NH * (HD / 2); base = k + (size_t)b * NKV * HD; head = pp >> 5; i = pp & 31; }
    const float inv_freq = __expf(-((float)(2 * i) / (float)HD) * 9.210340372f);
    const float ang = pos * inv_freq;
    const float cs = __cosf(ang), sn = __sinf(ang);
    float* e0 = base + (size_t)head * HD + i;
    float* e1 = e0 + HD / 2;
    const float x1 = *e0, x2 = *e1;
    *e0 = x1 * cs - x2 * sn;
    *e1 = x2 * cs + x1 * sn;
  }
}

// ---------------------------------------------------------------------------
// GQA flash-decode attention, chunked online softmax. grid = B*NKV, 256 thr.
// s==pos redirected to fresh k/v (paged insert without cache copy).
// ---------------------------------------------------------------------------
__global__ void attn_kernel(const float* __restrict__ q,
                            const float* __restrict__ knew,
                            const float* __restrict__ vnew,
                            const float* __restrict__ kc,
                            const float* __restrict__ vc,
                            const int* __restrict__ btab,
                            const int* __restrict__ seqlens,
                            float* __restrict__ attn_out) {
  __shared__ float sc[GQA][1024];
  __shared__ float qs[GQA][HD];
  __shared__ float red[256][GQA];
  __shared__ float accT[GQA][GQA][HD];

  const int t = threadIdx.x;
  const int b = blockIdx.x >> 3, kv = blockIdx.x & 7;
  const int pos = seqlens[b], S = pos + 1;
  const int sg = t >> 6, d = t & 63;
  const float scale = 0.125f;

  { int g = t >> 6, dd = t & 63;
    qs[g][dd] = q[(size_t)b * NH * HD + (kv * GQA + g) * HD + dd]; }
  __syncthreads();

  float m[GQA], l[GQA], acc[GQA];
  #pragma unroll
  for (int g = 0; g < GQA; ++g) { m[g] = -INFINITY; l[g] = 0.f; acc[g] = 0.f; }

  const float4* q0 = (const float4*)qs[0];
  const float4* q1 = (const float4*)qs[1];
  const float4* q2 = (const float4*)qs[2];
  const float4* q3 = (const float4*)qs[3];

  for (int c0 = 0; c0 < S; c0 += 1024) {
    const int nch = min(1024, S - c0);
    // ---- scores (b128 K-row loads; prefetch matching V row) ----
    for (int sl = t; sl < 1024; sl += 256) {
      float d0 = -INFINITY, d1 = -INFINITY, d2 = -INFINITY, d3 = -INFINITY;
      if (sl < nch) {
        const int s = c0 + sl;
        const float* krow;
        if (s == pos) krow = knew + (size_t)b * NKV * HD + kv * HD;
        else {
          const int blk = btab[b * NBLK + (s >> 4)];
          const size_t roff = ((size_t)(blk * BS + (s & 15)) * NKV + kv) * HD;
          krow = kc + roff;
          __builtin_prefetch(vc + roff, 0, 1);   // warm L2 for the V pass
        }
        const float4* kr = (const float4*)krow;
        d0 = d1 = d2 = d3 = 0.f;
        #pragma unroll
        for (int e = 0; e < HD / 4; ++e) {
          const float4 kk = kr[e];
          const float4 a0 = q0[e], a1 = q1[e], a2 = q2[e], a3 = q3[e];
          d0 += kk.x * a0.x + kk.y * a0.y + kk.z * a0.z + kk.w * a0.w;
          d1 += kk.x * a1.x + kk.y * a1.y + kk.z * a1.z + kk.w * a1.w;
          d2 += kk.x * a2.x + kk.y * a2.y + kk.z * a2.z + kk.w * a2.w;
          d3 += kk.x * a3.x + kk.y * a3.y + kk.z * a3.z + kk.w * a3.w;
        }
        d0 *= scale; d1 *= scale; d2 *= scale; d3 *= scale;
      }
      sc[0][sl] = d0; sc[1][sl] = d1; sc[2][sl] = d2; sc[3][sl] = d3;
    }
    __syncthreads();
    // ---- chunk max ----
    float lm[GQA];
    #pragma unroll
    for (int g = 0; g < GQA; ++g) lm[g] = -INFINITY;
    for (int sl = t; sl < 1024; sl += 256)
      #pragma unroll
      for (int g = 0; g < GQA; ++g) lm[g] = fmaxf(lm[g], sc[g][sl]);
    #pragma unroll
    for (int g = 0; g < GQA; ++g) red[t][g] = lm[g];
    __syncthreads();
    for (int off = 128; off > 0; off >>= 1) {
      if (t < off)
        #pragma unroll
        for (int g = 0; g < GQA; ++g)
          red[t][g] = fmaxf(red[t][g], red[t + off][g]);
      __syncthreads();
    }
    #pragma unroll
    for (int g = 0; g < GQA; ++g) {
      const float mnew = fmaxf(m[g], red[0][g]);
      const float alpha = (m[g] > -INFINITY) ? __expf(m[g] - mnew) : 0.f;
      acc[g] *= alpha; l[g] *= alpha; m[g] = mnew;
    }
    __syncthreads();
    // ---- p = exp(score - m), partial sums ----
    float ls[GQA];
    #pragma unroll
    for (int g = 0; g < GQA; ++g) ls[g] = 0.f;
    for (int sl = t; sl < 1024; sl += 256)
      #pragma unroll
      for (int g = 0; g < GQA; ++g) {
        const float p = __expf(sc[g][sl] - m[g]);
        sc[g][sl] = p; ls[g] += p;
      }
    #pragma unroll
    for (int g = 0; g < GQA; ++g) red[t][g] = ls[g];
    __syncthreads();
    for (int off = 128; off > 0; off >>= 1) {
      if (t < off)
        #pragma unroll
        for (int g = 0; g < GQA; ++g) red[t][g] += red[t + off][g];
      __syncthreads();
    }
    #pragma unroll
    for (int g = 0; g < GQA; ++g) l[g] += red[0][g];
    // ---- V accumulation: thread (sg,d), sl in [sg*256, sg*256+256) ----
    const int sbase = sg * 256;
    for (int j = 0; j < 256; ++j) {
      const int sl = sbase + j;
      if (sl >= nch) break;
      const int s = c0 + sl;
      float vv;
      if (s == pos) vv = vnew[(size_t)b * NKV * HD + kv * HD + d];
      else {
        const int blk = btab[b * NBLK + (s >> 4)];
        vv = vc[((size_t)(blk * BS + (s & 15)) * NKV + kv) * HD + d];
      }
      #pragma unroll
      for (int g = 0; g < GQA; ++g) acc[g] += sc[g][sl] * vv;
    }
    __syncthreads();
  }
  // ---- combine s-groups ----
  #pragma unroll
  for (int g = 0; g < GQA; ++g) accT[sg][g][d] = acc[g];
  __syncthreads();
  if (sg == 0) {
    #pragma unroll
    for (int g = 0; g < GQA; ++g) {
      const float tot = accT[0][g][d] + accT[1][g][d] + accT[2][g][d] + accT[3][g][d];
      attn_out[(size_t)b * NH * HD + (kv * GQA + g) * HD + d] = tot / l[g];
    }
  }
}

// ---------------------------------------------------------------------------
// O projection + residual: x1 = x + attn @ o_w
// ---------------------------------------------------------------------------
__global__ void oproj_kernel(const float* __restrict__ attn,
                             const float* __restrict__ ow,
                             const float* __restrict__ x,
                             float* __restrict__ x1) {
  __shared__ float wtile[2 * 512];
  const int tid = threadIdx.x;
  const int wave = tid >> 5, lane = tid & 31;
  const int lo = lane & 15, hi = lane >> 4;
  const int n0 = blockIdx.x * 16;
  const float* arow = attn + (size_t)(wave * 16 + lo) * HDIM + 2 * hi;
  v8f c = gemm_tile_lds(arow, ow + n0, HDIM, HDIM, wtile, tid, lo, hi);
  #pragma unroll
  for (int v = 0; v < 8; ++v) {
    const int m = wave * 16 + v + 8 * hi;
    x1[(size_t)m * HDIM + n0 + lo] = x[(size_t)m * HDIM + n0 + lo] + c[v];
  }
}

// ---------------------------------------------------------------------------
// Gate & up GEMMs + SiLU*up fused (shared A frag, LDS-staged dual B streams)
// ---------------------------------------------------------------------------
__global__ void gateup_kernel(const float* __restrict__ h,
                              const float* __restrict__ gw,
                              const float* __restrict__ uw,
                              float* __restrict__ act) {
  __shared__ float wtile[2 * 1024];   // [buf][mat(2)][32][16]
  const int tid = threadIdx.x;
  const int wave = tid >> 5, lane = tid & 31;
  const int lo = lane & 15, hi = lane >> 4;
  const int n0 = blockIdx.x * 16;
  const float* arow = h + (size_t)(wave * 16 + lo) * HDIM + 2 * hi;
  const float* G = gw + n0;
  const float* U = uw + n0;

  v8f cg0 = {}, cg1 = {}, cu0 = {}, cu1 = {};
  stage_tile(G, INTER, 0, wtile, tid);
  stage_tile(U, INTER, 0, wtile + 512, tid);
  for (int k0 = 0; k0 < HDIM; k0 += 32) {
    float* cur = wtile + (((k0 >> 5) & 1) << 10);
    float* nxt = wtile + ((((k0 >> 5) + 1) & 1) << 10);
    if (k0 + 32 < HDIM) {
      stage_tile(G, INTER, k0 + 32, nxt, tid);
      stage_tile(U, INTER, k0 + 32, nxt + 512, tid);
#ifdef USE_ASYNC_LDS
      __builtin_amdgcn_s_wait_asynccnt(4);
#endif
    } else {
#ifdef USE_ASYNC_LDS
      __builtin_amdgcn_s_wait_asynccnt(0);
#endif
    }
    __syncthreads();
    #pragma unroll
    for (int kk = 0; kk < 32; kk += 4) {
      v2f a  = { arow[k0 + kk], arow[k0 + kk + 1] };
      v2f bg = { cur[(kk + 2 * hi) * 16 + lo], cur[(kk + 2 * hi + 1) * 16 + lo] };
      v2f bu = { cur[512 + (kk + 2 * hi) * 16 + lo],
                 cur[512 + (kk + 2 * hi + 1) * 16 + lo] };
      if ((kk >> 2) & 1) {
        cg1 = __builtin_amdgcn_wmma_f32_16x16x4_f32(false, a, false, bg, (short)0, cg1, false, false);
        cu1 = __builtin_amdgcn_wmma_f32_16x16x4_f32(false, a, false, bu, (short)0, cu1, false, false);
      } else {
        cg0 = __builtin_amdgcn_wmma_f32_16x16x4_f32(false, a, false, bg, (short)0, cg0, false, false);
        cu0 = __builtin_amdgcn_wmma_f32_16x16x4_f32(false, a, false, bu, (short)0, cu0, false, false);
      }
    }
    __syncthreads();
  }
  const v8f cg = cg0 + cg1, cu = cu0 + cu1;
  #pragma unroll
  for (int v = 0; v < 8; ++v) {
    const int m = wave * 16 + v + 8 * hi;
    const float g = cg[v];
    const float silu = g / (1.0f + __expf(-g));
    act[(size_t)m * INTER + n0 + lo] = silu * cu[v];
  }
}

// ---------------------------------------------------------------------------
// Down GEMM + residual -> final output
// ---------------------------------------------------------------------------
__global__ void down_kernel(const float* __restrict__ act,
                            const float* __restrict__ dw,
                            const float* __restrict__ x1,
                            float* __restrict__ out) {
  __shared__ float wtile[2 * 512];
  const int tid = threadIdx.x;
  const int wave = tid >> 5, lane = tid & 31;
  const int lo = lane & 15, hi = lane >> 4;
  const int n0 = blockIdx.x * 16;
  const float* arow = act + (size_t)(wave * 16 + lo) * INTER + 2 * hi;
  v8f c = gemm_tile_lds(arow, dw + n0, INTER, HDIM, wtile, tid, lo, hi);
  #pragma unroll
  for (int v = 0; v < 8; ++v) {
    const int m = wave * 16 + v + 8 * hi;
    out[(size_t)m * HDIM + n0 + lo] = x1[(size_t)m * HDIM + n0 + lo] + c[v];
  }
}

// ---------------------------------------------------------------------------
// Host-side launcher
// ---------------------------------------------------------------------------
extern "C" void kernel_launch(void* const* d_in, const int* in_sizes, int n_in,
                              void* d_out, int out_size, void* d_ws, size_t ws_size,
                              hipStream_t stream) {
  const float* x        = (const float*)d_in[0];
  const float* k_cache  = (const float*)d_in[1];
  const float* v_cache  = (const float*)d_in[2];
  const float* ln1_w    = (const float*)d_in[3];
  const float* q_w      = (const float*)d_in[4];
  const float* k_w      = (const float*)d_in[5];
  const float* v_w      = (const float*)d_in[6];
  const float* o_w      = (const float*)d_in[7];
  const float* ln2_w    = (const float*)d_in[8];
  const float* gate_w   = (const float*)d_in[9];
  const float* up_w     = (const float*)d_in[10];
  const float* down_w   = (const float*)d_in[11];
  const int*   btab     = (const int*)d_in[12];
  const int*   seqlens  = (const int*)d_in[13];
  float* out = (float*)d_out;

  float* ws = (float*)d_ws;
  float* normed = ws;                 // 32*2048
  float* qbuf   = ws + 65536;         // 32*2048
  float* knew   = ws + 131072;        // 32*512
  float* vnew   = ws + 147456;        // 32*512
  float* attnb  = ws + 163840;        // 32*2048
  float* x1     = ws + 229376;        // 32*2048
  float* h2     = ws + 294912;        // 32*2048
  float* act    = ws + 360448;        // 32*8192

  rms_kernel   <<<BATCH, 256, 0, stream>>>(x, ln1_w, normed);
  qkv_kernel   <<<192,  64, 0, stream>>>(normed, q_w, k_w, v_w, qbuf, knew, vnew);
  rope_kernel  <<<BATCH, 256, 0, stream>>>(qbuf, knew, seqlens);
  attn_kernel  <<<BATCH * NKV, 256, 0, stream>>>(qbuf, knew, vnew, k_cache, v_cache,
                                                 btab, seqlens, attnb);
  oproj_kernel <<<HDIM / 16,  64, 0, stream>>>(attnb, o_w, x, x1);
  rms_kernel   <<<BATCH, 256, 0, stream>>>(x1, ln2_w, h2);
  gateup_kernel<<<INTER / 16, 64, 0, stream>>>(h2, gate_w, up_w, act);
  down_kernel  <<<HDIM / 16,  64, 0, stream>>>(act, down_w, x1, out);
}